// HOOffset_61735859913222
// MI455X (gfx1250) — compile-verified
//
#include <hip/hip_runtime.h>
#include <hip/hip_bf16.h>

// ---------------- problem constants ----------------
#define M_ANCH 704
#define N_ANCH 128
#define BATCH  64
#define NVERT  10475
#define NKB    10
#define DTOT   (M_ANCH*N_ANCH*3)   // 270336
#define NR     128                 // pca rank
#define K_OBJ  (N_ANCH*3)          // 384
#define K_SMP  (M_ANCH*3)          // 2112

// ---------------- workspace layout (float offsets, all even) ----------------
#define O_J0PART 0            // 16*33 = 528
#define O_J0C    544          // 33
#define O_SMPL   640          // 64*704*3 = 135168
#define O_OBJ    135808       // 64*128*3 = 24576
#define O_CENTER 160384       // 64*3
#define O_SSM    160576       // 64*3
#define O_MUS    160768       // 384
#define O_MC     161152       // 128
#define O_MCPART 161280       // 128*4
#define O_SMAT   161792       // 128*384 = 49152
#define O_SPART  210944       // 128*4*384 = 196608
#define O_SMM    407552       // 128*2112 = 270336
#define O_SVDM   677888       // 64*9
#define O_PMEAN  678464       // 64*3

typedef float v2f __attribute__((ext_vector_type(2)));
typedef float v4f __attribute__((ext_vector_type(4)));
typedef float v8f __attribute__((ext_vector_type(8)));

// =====================================================================
// K0: j0const reduction over all vertices:
//   a[c]      = sum_v Jreg0[v]*v_template[v,c]
//   bm[c][k]  = sum_v Jreg0[v]*shapedirs[v,c,k]
// 33 accumulators; 16 blocks of 256 threads -> partials
// =====================================================================
__global__ __launch_bounds__(256) void kJ0Part(const float* __restrict__ jreg,
                                               const float* __restrict__ vt,
                                               const float* __restrict__ sd,
                                               float* __restrict__ part) {
    __shared__ float lds[256 * 33];
    const int tid = threadIdx.x;
    float acc[33];
#pragma unroll
    for (int j = 0; j < 33; ++j) acc[j] = 0.f;
    for (int v = blockIdx.x * 256 + tid; v < NVERT; v += 16 * 256) {
        float jv = jreg[v];               // row 0 of J_regressor
        const float* vtp = vt + v * 3;
        const float* sdp = sd + v * 30;
#pragma unroll
        for (int c = 0; c < 3; ++c) acc[c] += jv * vtp[c];
#pragma unroll
        for (int j = 0; j < 30; ++j) acc[3 + j] += jv * sdp[j];
    }
#pragma unroll
    for (int j = 0; j < 33; ++j) lds[tid * 33 + j] = acc[j];
    __syncthreads();
    for (int off = 128; off >= 1; off >>= 1) {
        if (tid < off) {
#pragma unroll
            for (int j = 0; j < 33; ++j) lds[tid * 33 + j] += lds[(tid + off) * 33 + j];
        }
        __syncthreads();
    }
    if (tid < 33) part[blockIdx.x * 33 + tid] = lds[tid];
}

__global__ __launch_bounds__(64) void kJ0Comb(const float* __restrict__ part,
                                              float* __restrict__ j0c) {
    int j = threadIdx.x;
    if (j < 33) {
        float s = 0.f;
#pragma unroll
        for (int p = 0; p < 16; ++p) s += part[p * 33 + j];
        j0c[j] = s;
    }
}

// =====================================================================
// K1: smpl_anchors[b,m,c] = vt[ai,c] + sum_k beta*sd[ai,c,k] - J0[b,c]
// =====================================================================
__global__ __launch_bounds__(256) void kSmpl(const float* __restrict__ betas,
                                             const float* __restrict__ vt,
                                             const float* __restrict__ sd,
                                             const int* __restrict__ aidx,
                                             const float* __restrict__ j0c,
                                             float* __restrict__ smpl) {
    int gid = blockIdx.x * 256 + threadIdx.x;        // exactly 64*704
    int b = gid / M_ANCH, m = gid % M_ANCH;
    if (b >= BATCH) return;
    float bt[NKB];
#pragma unroll
    for (int k = 0; k < NKB; ++k) bt[k] = betas[b * NKB + k];
    int ai = aidx[m];
    const float* vtp = vt + ai * 3;
    const float* sdp = sd + ai * 30;
    float* out = smpl + (size_t)gid * 3;
#pragma unroll
    for (int c = 0; c < 3; ++c) {
        float j0 = j0c[c];
        float vv = vtp[c];
#pragma unroll
        for (int k = 0; k < NKB; ++k) {
            j0 += j0c[3 + c * 10 + k] * bt[k];
            vv += sdp[c * 10 + k] * bt[k];
        }
        out[c] = vv - j0;
    }
}

// =====================================================================
// K2: obj_anchors[b,n,d] = sum_c obj_org[b,n,c]*rot[b,d,c] + trans[b,d]
//     centerQ[b,d] = mean_n obj_anchors (== mean of broadcast Q)
// =====================================================================
__global__ __launch_bounds__(128) void kObj(const float* __restrict__ org,
                                            const float* __restrict__ rot,
                                            const float* __restrict__ trans,
                                            float* __restrict__ obj,
                                            float* __restrict__ center) {
    __shared__ float red[128 * 3];
    int b = blockIdx.x, n = threadIdx.x;
    const float* o = org + (size_t)(b * N_ANCH + n) * 3;
    const float* rr = rot + b * 9;
    float out[3];
#pragma unroll
    for (int d = 0; d < 3; ++d) {
        out[d] = trans[b * 3 + d] + o[0] * rr[d * 3 + 0] + o[1] * rr[d * 3 + 1] + o[2] * rr[d * 3 + 2];
        obj[(size_t)(b * N_ANCH + n) * 3 + d] = out[d];
        red[n * 3 + d] = out[d];
    }
    __syncthreads();
    for (int off = 64; off >= 1; off >>= 1) {
        if (n < off) {
#pragma unroll
            for (int d = 0; d < 3; ++d) red[n * 3 + d] += red[(n + off) * 3 + d];
        }
        __syncthreads();
    }
    if (n < 3) center[b * 3 + n] = red[n] * (1.f / N_ANCH);
}

// =====================================================================
// K_ssm: Ssm[b,c] = sum_m smpl_anchors[b,m,c]
// =====================================================================
__global__ __launch_bounds__(256) void kSsm(const float* __restrict__ smpl,
                                            float* __restrict__ ssm) {
    __shared__ float red[256 * 3];
    int b = blockIdx.x, tid = threadIdx.x;
    float a0 = 0.f, a1 = 0.f, a2 = 0.f;
    for (int m = tid; m < M_ANCH; m += 256) {
        const float* p = smpl + (size_t)(b * M_ANCH + m) * 3;
        a0 += p[0]; a1 += p[1]; a2 += p[2];
    }
    red[tid * 3 + 0] = a0; red[tid * 3 + 1] = a1; red[tid * 3 + 2] = a2;
    __syncthreads();
    for (int off = 128; off >= 1; off >>= 1) {
        if (tid < off) {
#pragma unroll
            for (int d = 0; d < 3; ++d) red[tid * 3 + d] += red[(tid + off) * 3 + d];
        }
        __syncthreads();
    }
    if (tid < 3) ssm[b * 3 + tid] = red[tid];
}

// =====================================================================
// K_mu: mu_s[j=(n*3+c)] = sum_m pca_mean[m*384 + j]   (coalesced over j)
// =====================================================================
__global__ __launch_bounds__(384) void kMu(const float* __restrict__ mean,
                                           float* __restrict__ mus) {
    int j = threadIdx.x;
    float a = 0.f;
    for (int m = 0; m < M_ANCH; ++m) a += mean[m * 384 + j];
    mus[j] = a;
}

// =====================================================================
// K_C: the single streaming pass over pca_components (138 MB).
// grid (128 r, 4 msplit), 256 thr (8 waves). Wave handles 22 m-blocks
// (each m-block = 384 consecutive floats). Lane owns 12 consecutive
// floats (48B) -> fixed (n,c) slots, c = t%3 compile-time.
// Produces: Spart[r,ms,384], Sm[r,m,3] (direct), mcPart[r,ms].
// =====================================================================
__global__ __launch_bounds__(256) void kReduceC(const float* __restrict__ C,
                                                const float* __restrict__ mean,
                                                float* __restrict__ spart,
                                                float* __restrict__ smm,
                                                float* __restrict__ mcpart) {
    __shared__ float lds[8 * 384 + 8];
    const int r = blockIdx.x, ms = blockIdx.y;
    const int tid = threadIdx.x, w = tid >> 5, lane = tid & 31;
    const float* Crow = C + (size_t)r * DTOT;
    float sacc[12];
#pragma unroll
    for (int t = 0; t < 12; ++t) sacc[t] = 0.f;
    float mcacc = 0.f;

    for (int mi = 0; mi < 22; ++mi) {
        const int m = ms * 176 + w + mi * 8;
        const float* p = Crow + m * 384 + lane * 12;
        const float* q = mean + m * 384 + lane * 12;
        v4f c0 = __builtin_nontemporal_load((const v4f*)(p));
        v4f c1 = __builtin_nontemporal_load((const v4f*)(p + 4));
        v4f c2 = __builtin_nontemporal_load((const v4f*)(p + 8));
        v4f m0 = *(const v4f*)(q);
        v4f m1 = *(const v4f*)(q + 4);
        v4f m2 = *(const v4f*)(q + 8);
        // S accumulation (slot j = lane*12 + t)
        sacc[0] += c0.x; sacc[1] += c0.y; sacc[2]  += c0.z; sacc[3]  += c0.w;
        sacc[4] += c1.x; sacc[5] += c1.y; sacc[6]  += c1.z; sacc[7]  += c1.w;
        sacc[8] += c2.x; sacc[9] += c2.y; sacc[10] += c2.z; sacc[11] += c2.w;
        // mc = <mean, C_row>
        mcacc += c0.x * m0.x + c0.y * m0.y + c0.z * m0.z + c0.w * m0.w
               + c1.x * m1.x + c1.y * m1.y + c1.z * m1.z + c1.w * m1.w
               + c2.x * m2.x + c2.y * m2.y + c2.z * m2.z + c2.w * m2.w;
        // Sm per-c partial (c = t%3, t = local index 0..11)
        float p0 = c0.x + c0.w + c1.z + c2.y;   // t = 0,3,6,9
        float p1 = c0.y + c1.x + c1.w + c2.z;   // t = 1,4,7,10
        float p2 = c0.z + c1.y + c2.x + c2.w;   // t = 2,5,8,11
#pragma unroll
        for (int off = 16; off >= 1; off >>= 1) {
            p0 += __shfl_xor(p0, off, 32);
            p1 += __shfl_xor(p1, off, 32);
            p2 += __shfl_xor(p2, off, 32);
        }
        if (lane == 0) {
            float* d = smm + (size_t)r * K_SMP + m * 3;
            d[0] = p0; d[1] = p1; d[2] = p2;
        }
    }
#pragma unroll
    for (int t = 0; t < 12; ++t) lds[w * 384 + lane * 12 + t] = sacc[t];
#pragma unroll
    for (int off = 16; off >= 1; off >>= 1) mcacc += __shfl_xor(mcacc, off, 32);
    if (lane == 0) lds[8 * 384 + w] = mcacc;
    __syncthreads();
    for (int j = tid; j < 384; j += 256) {
        float s = 0.f;
#pragma unroll
        for (int ww = 0; ww < 8; ++ww) s += lds[ww * 384 + j];
        spart[(size_t)(r * 4 + ms) * 384 + j] = s;
    }
    if (tid == 0) {
        float s = 0.f;
#pragma unroll
        for (int ww = 0; ww < 8; ++ww) s += lds[8 * 384 + ww];
        mcpart[r * 4 + ms] = s;
    }
}

// combine msplit partials: S[r,384], mc[r]
__global__ __launch_bounds__(256) void kCombS(const float* __restrict__ spart,
                                              const float* __restrict__ mcpart,
                                              float* __restrict__ S,
                                              float* __restrict__ mc) {
    int r = blockIdx.x, tid = threadIdx.x;
    for (int j = tid; j < 384; j += 256) {
        float s = 0.f;
#pragma unroll
        for (int msi = 0; msi < 4; ++msi) s += spart[(size_t)(r * 4 + msi) * 384 + j];
        S[(size_t)r * 384 + j] = s;
    }
    if (tid == 0) {
        float s = 0.f;
#pragma unroll
        for (int msi = 0; msi < 4; ++msi) s += mcpart[r * 4 + msi];
        mc[r] = s;
    }
}

// =====================================================================
// kGamma: gamma[b,r] = <obj_b, S_r>_384 - <smpl_b, Sm_r>_2112 - mc[r]
// WMMA f32 16x16x4. Block = 8 waves, wave -> r-tile; blockIdx -> b-tile.
// A frag (16x4): lane L<16: rows, K=k,k+1 ; L>=16: K=k+2,k+3  -> float2
// B frag (4x16): lane L<16: cols, K=k,k+1 ; L>=16: K=k+2,k+3  -> float2
// =====================================================================
__global__ __launch_bounds__(256) void kGamma(const float* __restrict__ obj,
                                              const float* __restrict__ smpl,
                                              const float* __restrict__ S,
                                              const float* __restrict__ Sm,
                                              const float* __restrict__ mc,
                                              float* __restrict__ gout) {
    const int lane = threadIdx.x & 31;
    const int wv = threadIdx.x >> 5;
    const int b0 = blockIdx.x * 16;
    const int r0 = wv * 16;
    const int row = lane & 15;
    const int koff = (lane >> 4) * 2;

    v8f acc1 = {0.f, 0.f, 0.f, 0.f, 0.f, 0.f, 0.f, 0.f};
    v8f acc2 = {0.f, 0.f, 0.f, 0.f, 0.f, 0.f, 0.f, 0.f};

    {
        const float* Ap = obj + (size_t)(b0 + row) * K_OBJ + koff;
        const float* Bp = S + (size_t)(r0 + row) * K_OBJ + koff;
#pragma unroll 4
        for (int k = 0; k < K_OBJ; k += 4) {
            v2f a = *(const v2f*)(Ap + k);
            v2f bb = *(const v2f*)(Bp + k);
            acc1 = __builtin_amdgcn_wmma_f32_16x16x4_f32(false, a, false, bb,
                                                         (short)0, acc1, false, false);
        }
    }
    {
        const float* Ap = smpl + (size_t)(b0 + row) * K_SMP + koff;
        const float* Bp = Sm + (size_t)(r0 + row) * K_SMP + koff;
#pragma unroll 4
        for (int k = 0; k < K_SMP; k += 4) {
            v2f a = *(const v2f*)(Ap + k);
            v2f bb = *(const v2f*)(Bp + k);
            acc2 = __builtin_amdgcn_wmma_f32_16x16x4_f32(false, a, false, bb,
                                                         (short)0, acc2, false, false);
        }
    }
    const int col = r0 + row;
    const float mcv = mc[col];
#pragma unroll
    for (int i = 0; i < 8; ++i) {
        const int mrow = (lane < 16) ? i : (i + 8);
        gout[(size_t)(b0 + mrow) * NR + col] = acc1[i] - acc2[i] - mcv;
    }
}

// =====================================================================
// kSvdMat: per batch:
//   PS[n,c] = Ssm[b,c] + mu_s[n,c] + sum_r gamma[b,r]*S[r,n,c]
//   svd_mat[b,c,d] = sum_n PS[n,c]*(obj[b,n,d]-center[b,d])
//   Pmean[b,c]     = (sum_n PS[n,c]) / (704*128)
// =====================================================================
__global__ __launch_bounds__(128) void kSvdMat(const float* __restrict__ obj,
                                               const float* __restrict__ center,
                                               const float* __restrict__ ssm,
                                               const float* __restrict__ mus,
                                               const float* __restrict__ S,
                                               const float* __restrict__ gamma,
                                               float* __restrict__ svdm,
                                               float* __restrict__ pmean) {
    __shared__ float g[NR];
    __shared__ float red[128 * 12];
    const int b = blockIdx.x, n = threadIdx.x;
    g[n] = gamma[b * NR + n];
    __syncthreads();
    const float cx = center[b * 3 + 0], cy = center[b * 3 + 1], cz = center[b * 3 + 2];
    const float* o = obj + (size_t)(b * N_ANCH + n) * 3;
    const float q0 = o[0] - cx, q1 = o[1] - cy, q2 = o[2] - cz;
    float ps0 = ssm[b * 3 + 0] + mus[n * 3 + 0];
    float ps1 = ssm[b * 3 + 1] + mus[n * 3 + 1];
    float ps2 = ssm[b * 3 + 2] + mus[n * 3 + 2];
    const float* Sp = S + n * 3;
    for (int r = 0; r < NR; ++r) {
        const float gv = g[r];
        ps0 += gv * Sp[(size_t)r * 384 + 0];
        ps1 += gv * Sp[(size_t)r * 384 + 1];
        ps2 += gv * Sp[(size_t)r * 384 + 2];
    }
    float* my = red + n * 12;
    my[0] = ps0 * q0; my[1] = ps0 * q1; my[2] = ps0 * q2;
    my[3] = ps1 * q0; my[4] = ps1 * q1; my[5] = ps1 * q2;
    my[6] = ps2 * q0; my[7] = ps2 * q1; my[8] = ps2 * q2;
    my[9] = ps0; my[10] = ps1; my[11] = ps2;
    __syncthreads();
    for (int off = 64; off >= 1; off >>= 1) {
        if (n < off) {
#pragma unroll
            for (int j = 0; j < 12; ++j) red[n * 12 + j] += red[(n + off) * 12 + j];
        }
        __syncthreads();
    }
    if (n == 0) {
#pragma unroll
        for (int j = 0; j < 9; ++j) svdm[b * 9 + j] = red[j];
#pragma unroll
        for (int d = 0; d < 3; ++d)
            pmean[b * 3 + d] = red[9 + d] * (1.f / (float)(M_ANCH * N_ANCH));
    }
}

// =====================================================================
// kSvd: 3x3 SVD (Jacobi on M^T M) + Kabsch:  R, T.  One thread per batch.
// =====================================================================
__global__ __launch_bounds__(64) void kSvd(const float* __restrict__ svdm,
                                           const float* __restrict__ pmean,
                                           const float* __restrict__ center,
                                           float* __restrict__ Rout,
                                           float* __restrict__ Tout) {
    const int b = threadIdx.x;
    if (b >= BATCH) return;
    float Mm[3][3], A[3][3], Vv[3][3];
#pragma unroll
    for (int i = 0; i < 3; ++i)
#pragma unroll
        for (int j = 0; j < 3; ++j) Mm[i][j] = svdm[b * 9 + i * 3 + j];
#pragma unroll
    for (int i = 0; i < 3; ++i)
#pragma unroll
        for (int j = 0; j < 3; ++j) {
            float s = 0.f;
#pragma unroll
            for (int k = 0; k < 3; ++k) s += Mm[k][i] * Mm[k][j];
            A[i][j] = s;
            Vv[i][j] = (i == j) ? 1.f : 0.f;
        }
    const int PP[3] = {0, 0, 1}, QQ[3] = {1, 2, 2};
    for (int sweep = 0; sweep < 8; ++sweep) {
        for (int pair = 0; pair < 3; ++pair) {
            const int p = PP[pair], q = QQ[pair];
            const float apq = A[p][q];
            if (fabsf(apq) > 1e-18f) {
                const float tau = (A[q][q] - A[p][p]) / (2.f * apq);
                const float t = ((tau >= 0.f) ? 1.f : -1.f) / (fabsf(tau) + sqrtf(1.f + tau * tau));
                const float c = 1.f / sqrtf(1.f + t * t);
                const float s = t * c;
                for (int k = 0; k < 3; ++k) {
                    const float akp = A[k][p], akq = A[k][q];
                    A[k][p] = c * akp - s * akq;
                    A[k][q] = s * akp + c * akq;
                }
                for (int k = 0; k < 3; ++k) {
                    const float apk = A[p][k], aqk = A[q][k];
                    A[p][k] = c * apk - s * aqk;
                    A[q][k] = s * apk + c * aqk;
                }
                for (int k = 0; k < 3; ++k) {
                    const float vkp = Vv[k][p], vkq = Vv[k][q];
                    Vv[k][p] = c * vkp - s * vkq;
                    Vv[k][q] = s * vkp + c * vkq;
                }
            }
        }
    }
    float lam[3] = {A[0][0], A[1][1], A[2][2]};
    // sort descending, permute V columns
    for (int i = 0; i < 2; ++i)
        for (int j = i + 1; j < 3; ++j)
            if (lam[j] > lam[i]) {
                float tl = lam[i]; lam[i] = lam[j]; lam[j] = tl;
                for (int k = 0; k < 3; ++k) {
                    float tv = Vv[k][i]; Vv[k][i] = Vv[k][j]; Vv[k][j] = tv;
                }
            }
    float U[3][3];
#pragma unroll
    for (int i = 0; i < 3; ++i) {
        float sv = sqrtf(fmaxf(lam[i], 0.f));
        float inv = (sv > 1e-12f) ? (1.f / sv) : 0.f;
        for (int k = 0; k < 3; ++k) {
            float s = 0.f;
            for (int j = 0; j < 3; ++j) s += Mm[k][j] * Vv[j][i];
            U[k][i] = s * inv;
        }
    }
    auto det3 = [](const float X[3][3]) {
        return X[0][0] * (X[1][1] * X[2][2] - X[1][2] * X[2][1])
             - X[0][1] * (X[1][0] * X[2][2] - X[1][2] * X[2][0])
             + X[0][2] * (X[1][0] * X[2][1] - X[1][1] * X[2][0]);
    };
    const float d = det3(U) * det3(Vv);   // det(u @ vh)
    float dv[3] = {1.f, 1.f, d};
    float R[3][3];
#pragma unroll
    for (int i = 0; i < 3; ++i)
#pragma unroll
        for (int j = 0; j < 3; ++j) {
            float s = 0.f;
            for (int k = 0; k < 3; ++k) s += U[i][k] * dv[k] * Vv[j][k]; // vh = V^T
            R[i][j] = s;
            Rout[b * 9 + i * 3 + j] = s;
        }
#pragma unroll
    for (int c = 0; c < 3; ++c) {
        float t = pmean[b * 3 + c];
        for (int j = 0; j < 3; ++j) t -= R[c][j] * center[b * 3 + j];
        Tout[b * 3 + c] = t;
    }
}

// =====================================================================
extern "C" void kernel_launch(void* const* d_in, const int* in_sizes, int n_in,
                              void* d_out, int out_size, void* d_ws, size_t ws_size,
                              hipStream_t stream) {
    (void)in_sizes; (void)n_in; (void)out_size; (void)ws_size;
    const float* betas = (const float*)d_in[0];
    const float* rot   = (const float*)d_in[1];
    const float* trans = (const float*)d_in[2];
    const float* objorg= (const float*)d_in[3];
    const float* vt    = (const float*)d_in[4];
    const float* sd    = (const float*)d_in[5];
    const float* jreg  = (const float*)d_in[6];
    const float* pmean = (const float*)d_in[7];
    const float* pcomp = (const float*)d_in[8];
    const int*   aidx  = (const int*)d_in[9];
    float* ws  = (float*)d_ws;
    float* out = (float*)d_out;
    float* gamma_out = out;                 // 64*128
    float* Rout = out + BATCH * NR;         // 64*9
    float* Tout = Rout + BATCH * 9;         // 64*3

    kJ0Part<<<16, 256, 0, stream>>>(jreg, vt, sd, ws + O_J0PART);
    kJ0Comb<<<1, 64, 0, stream>>>(ws + O_J0PART, ws + O_J0C);
    kSmpl<<<(BATCH * M_ANCH) / 256, 256, 0, stream>>>(betas, vt, sd, aidx,
                                                      ws + O_J0C, ws + O_SMPL);
    kObj<<<BATCH, 128, 0, stream>>>(objorg, rot, trans, ws + O_OBJ, ws + O_CENTER);
    kSsm<<<BATCH, 256, 0, stream>>>(ws + O_SMPL, ws + O_SSM);
    kMu<<<1, 384, 0, stream>>>(pmean, ws + O_MUS);
    dim3 gC(NR, 4);
    kReduceC<<<gC, 256, 0, stream>>>(pcomp, pmean, ws + O_SPART, ws + O_SMM, ws + O_MCPART);
    kCombS<<<NR, 256, 0, stream>>>(ws + O_SPART, ws + O_MCPART, ws + O_SMAT, ws + O_MC);
    kGamma<<<BATCH / 16, 256, 0, stream>>>(ws + O_OBJ, ws + O_SMPL, ws + O_SMAT,
                                           ws + O_SMM, ws + O_MC, gamma_out);
    kSvdMat<<<BATCH, 128, 0, stream>>>(ws + O_OBJ, ws + O_CENTER, ws + O_SSM,
                                       ws + O_MUS, ws + O_SMAT, gamma_out,
                                       ws + O_SVDM, ws + O_PMEAN);
    kSvd<<<1, 64, 0, stream>>>(ws + O_SVDM, ws + O_PMEAN, ws + O_CENTER, Rout, Tout);
}